// Multi_head_ltee_2216203124967
// MI455X (gfx1250) — compile-verified
//
#include <hip/hip_runtime.h>
#include <math.h>

#define H      50
#define TSN    7
#define NE     5
#define NB     4096
#define RSTRIDE 64        // padded row stride of rep buffer (floats)
#define LDM    4112       // padded leading dim of distance matrix (16*257 floats)
#define WPAD   (160 * 52) // padded gate-weight image per expert (floats)

typedef __attribute__((ext_vector_type(2))) float v2f;
typedef __attribute__((ext_vector_type(8))) float v8f;

// ---------------------------------------------------------------------------
// CDNA5 WMMA helper: D = A(16x4,f32) * B(4x16,f32) + C(16x16,f32)
// ---------------------------------------------------------------------------
__device__ __forceinline__ v8f wmma4(v2f a, v2f b, v8f c) {
  return __builtin_amdgcn_wmma_f32_16x16x4_f32(
      /*neg_a=*/false, a, /*neg_b=*/false, b,
      /*c_mod=*/(short)0, c, /*reuse_a=*/false, /*reuse_b=*/false);
}

__device__ __forceinline__ float wave_sum(float v) {
#pragma unroll
  for (int off = 16; off > 0; off >>= 1) v += __shfl_xor(v, off, 32);
  return v;
}
__device__ __forceinline__ float wave_max(float v) {
#pragma unroll
  for (int off = 16; off > 0; off >>= 1) v = fmaxf(v, __shfl_xor(v, off, 32));
  return v;
}
__device__ __forceinline__ float sigmoidf_(float x) {
  return 1.0f / (1.0f + __expf(-x));
}

// ---------------------------------------------------------------------------
// Gate GEMM: sOut[16x160] = sA[16x52] @ Wp^T + bp
// Wp is the PRE-PADDED (160x52, zero-filled) weight image -> unconditional
// 8B-aligned b64 loads, no EXEC-predicated guards around the WMMA stream.
// A-fragments are hoisted (invariant over the 10 N-tiles).
// ---------------------------------------------------------------------------
__device__ void gate_gemm(const float* __restrict__ sA,
                          const float* __restrict__ Wp,
                          const float* __restrict__ bp,
                          float* __restrict__ sOut, int lane) {
  const int half = lane >> 4, lm = lane & 15;
  v2f a[13];
#pragma unroll
  for (int kc = 0; kc < 13; ++kc)
    a[kc] = *(const v2f*)&sA[lm * 52 + kc * 4 + 2 * half];
  for (int nt = 0; nt < 10; ++nt) {
    const int nn = nt * 16 + lm;
    const float* wrow = Wp + nn * 52 + 2 * half;
    v8f c = {0.f, 0.f, 0.f, 0.f, 0.f, 0.f, 0.f, 0.f};
#pragma unroll
    for (int kc = 0; kc < 13; ++kc) {
      v2f b = *(const v2f*)(wrow + kc * 4);
      c = wmma4(a[kc], b, c);
    }
    const float bb = bp[nn];
#pragma unroll
    for (int r = 0; r < 8; ++r) sOut[(r + 8 * half) * 160 + nn] = c[r] + bb;
  }
}

// ---------------------------------------------------------------------------
// Pad gate weights/biases into zero-filled (E,160,52)/(E,160) images.
// ---------------------------------------------------------------------------
__global__ void pad_w_kernel(const float* __restrict__ W_ih,
                             const float* __restrict__ W_hh,
                             const float* __restrict__ b_ih,
                             const float* __restrict__ b_hh,
                             float* __restrict__ WihP, float* __restrict__ WhhP,
                             float* __restrict__ bihP,
                             float* __restrict__ bhhP) {
  const int idx = blockIdx.x * 256 + threadIdx.x;
  if (idx < NE * WPAD) {
    const int e = idx / WPAD;
    const int r = (idx / 52) % 160;
    const int k = idx % 52;
    const bool in = (r < 3 * H) && (k < H);
    WihP[idx] = in ? W_ih[(size_t)e * 3 * H * H + r * H + k] : 0.f;
    WhhP[idx] = in ? W_hh[(size_t)e * 3 * H * H + r * H + k] : 0.f;
  }
  if (idx < NE * 160) {
    const int e = idx / 160, r = idx % 160;
    bihP[idx] = (r < 3 * H) ? b_ih[e * 3 * H + r] : 0.f;
    bhhP[idx] = (r < 3 * H) ? b_hh[e * 3 * H + r] : 0.f;
  }
}

// ---------------------------------------------------------------------------
// Stage 1: per-expert GRU + heads. Block = 1 wave, 16 batch rows; masked
// writes (each row belongs to exactly one expert bucket).
// ---------------------------------------------------------------------------
__global__ __launch_bounds__(32) void expert_kernel(
    const float* __restrict__ x, const float* __restrict__ WihP,
    const float* __restrict__ WhhP, const float* __restrict__ bihP,
    const float* __restrict__ bhhP, const float* __restrict__ Wfc,
    const float* __restrict__ bfc, const float* __restrict__ u_att,
    const float* __restrict__ Wt, const float* __restrict__ bt,
    const float* __restrict__ Ws1, const float* __restrict__ bs1,
    const float* __restrict__ Ws2, const float* __restrict__ bs2,
    const float* __restrict__ Wl1, const float* __restrict__ bl1,
    const float* __restrict__ Wl2, const float* __restrict__ bl2,
    float* __restrict__ Rpad, float* __restrict__ rnrm,
    float* __restrict__ out_s, float* __restrict__ out_y) {
  __shared__ float sF[16 * 52];
  __shared__ float sH[16 * 52];
  __shared__ float sGIGH[2 * 16 * 160];  // gi | gh; reused as pre|tr later
  __shared__ float sRNN[TSN * 16 * 52];
  __shared__ float sTreat[16 * 52];
  __shared__ float sSim[16 * 8];
  __shared__ float sW[16 * 8];
  __shared__ float sT[16];
  __shared__ int sMask[16];

  float* sGI = sGIGH;
  float* sGH = sGIGH + 16 * 160;

  const int lane = threadIdx.x;
  const int e = blockIdx.y;
  const int r0 = blockIdx.x * 16;

  // --- load feat (padded K), t, mask; zero hidden state ---
  for (int q = lane; q < 16 * 52; q += 32) {
    const int m = q / 52, k = q % 52;
    sF[q] = (k < H) ? x[(size_t)(r0 + m) * (H + 1) + 1 + k] : 0.f;
    sH[q] = 0.f;
  }
  if (lane < 16) {
    const float tv = x[(size_t)(r0 + lane) * (H + 1)];
    sT[lane] = tv;
    const float lo = 0.2f * e, hi = 0.2f * (e + 1);
    sMask[lane] = (tv >= lo) && ((e == NE - 1) ? (tv <= hi) : (tv < hi));
  }
  __syncthreads();

  // --- gi = feat @ W_ih^T + b_ih (WMMA) ---
  gate_gemm(sF, WihP + (size_t)e * WPAD, bihP + e * 160, sGI, lane);
  __syncthreads();

  // --- GRU: 7 steps ---
  for (int st = 0; st < TSN; ++st) {
    gate_gemm(sH, WhhP + (size_t)e * WPAD, bhhP + e * 160, sGH, lane);
    __syncthreads();
    float hn[25];
    int c2 = 0;
    for (int q = lane; q < 16 * H; q += 32, ++c2) {
      const int m = q / H, d = q % H;
      const float r = sigmoidf_(sGI[m * 160 + d] + sGH[m * 160 + d]);
      const float z = sigmoidf_(sGI[m * 160 + H + d] + sGH[m * 160 + H + d]);
      const float nn =
          tanhf(sGI[m * 160 + 2 * H + d] + r * sGH[m * 160 + 2 * H + d]);
      hn[c2] = (1.f - z) * nn + z * sH[m * 52 + d];
    }
    __syncthreads();
    c2 = 0;
    for (int q = lane; q < 16 * H; q += 32, ++c2) {
      const int m = q / H, d = q % H;
      sH[m * 52 + d] = hn[c2];
      sRNN[(st * 16 + m) * 52 + d] = hn[c2];
    }
    __syncthreads();
  }

  // --- treat = t*Wt + bt ---
  for (int q = lane; q < 16 * 52; q += 32) {
    const int m = q / 52, k = q % 52;
    sTreat[q] = (k < H) ? (sT[m] * Wt[e * H + k] + bt[e * H + k]) : 0.f;
  }
  __syncthreads();

  float* sPre = sGIGH;        // 16x200
  float* sTr = sGIGH + 3200;  // 16x52
  float* sCtx = sF;           // reuse feat space

  // --- per-timestep short head + attention logits ---
  for (int st = 0; st < TSN; ++st) {
    for (int q = lane; q < 16 * 200; q += 32) {
      const int m = q / 200, o = q % 200;
      const float* w1 = Ws1 + (size_t)e * 200 * 100 + o * 100;
      float acc = bs1[e * 200 + o];
      for (int k = 0; k < H; ++k) acc += sRNN[(st * 16 + m) * 52 + k] * w1[k];
      for (int k = 0; k < H; ++k) acc += sTreat[m * 52 + k] * w1[H + k];
      sPre[m * 200 + o] = fmaxf(acc, 0.f);
    }
    for (int q = lane; q < 16 * H; q += 32) {
      const int m = q / H, d = q % H;
      const float* wf = Wfc + (size_t)e * H * H + d * H;
      float acc = bfc[e * H + d];
      for (int k = 0; k < H; ++k) acc += sRNN[(st * 16 + m) * 52 + k] * wf[k];
      sTr[m * 52 + d] = tanhf(acc);
    }
    __syncthreads();
    if (lane < 16) {
      const int m = lane;
      float acc = bs2[e];
      for (int o = 0; o < 200; ++o) acc += sPre[m * 200 + o] * Ws2[e * 200 + o];
      if (sMask[m]) out_s[(size_t)(r0 + m) * TSN + st] = acc;
      float s = 0.f;
      for (int d = 0; d < H; ++d) s += sTr[m * 52 + d] * u_att[e * H + d];
      sSim[m * 8 + st] = s;
    }
    __syncthreads();
  }

  // --- softmax over timesteps ---
  if (lane < 16) {
    float mx = -1e30f;
    for (int st = 0; st < TSN; ++st) mx = fmaxf(mx, sSim[lane * 8 + st]);
    float s = 0.f, wv[TSN];
    for (int st = 0; st < TSN; ++st) {
      wv[st] = __expf(sSim[lane * 8 + st] - mx);
      s += wv[st];
    }
    for (int st = 0; st < TSN; ++st) sW[lane * 8 + st] = wv[st] / s;
  }
  __syncthreads();

  // --- ctx = sum_ts w * tanh(fc(h_ts)) (recompute transformed) ---
  for (int q = lane; q < 16 * H; q += 32) {
    const int m = q / H, d = q % H;
    const float* wf = Wfc + (size_t)e * H * H + d * H;
    const float bb = bfc[e * H + d];
    float acc = 0.f;
    for (int st = 0; st < TSN; ++st) {
      float a2 = bb;
      for (int k = 0; k < H; ++k) a2 += sRNN[(st * 16 + m) * 52 + k] * wf[k];
      acc += sW[m * 8 + st] * tanhf(a2);
    }
    sCtx[m * 52 + d] = acc;
  }
  __syncthreads();

  // --- long head ---
  for (int q = lane; q < 16 * 200; q += 32) {
    const int m = q / 200, o = q % 200;
    const float* w1 = Wl1 + (size_t)e * 200 * 100 + o * 100;
    float acc = bl1[e * 200 + o];
    for (int k = 0; k < H; ++k) acc += sCtx[m * 52 + k] * w1[k];
    for (int k = 0; k < H; ++k) acc += sTreat[m * 52 + k] * w1[H + k];
    sPre[m * 200 + o] = fmaxf(acc, 0.f);
  }
  __syncthreads();
  if (lane < 16) {
    float acc = bl2[e];
    for (int o = 0; o < 200; ++o) acc += sPre[lane * 200 + o] * Wl2[e * 200 + o];
    if (sMask[lane]) out_y[r0 + lane] = acc;
  }

  // --- masked write of rep (padded to 52 zero cols) + row norms ---
  for (int st = 0; st < TSN; ++st) {
    for (int q = lane; q < 16 * 52; q += 32) {
      const int m = q / 52, k = q % 52;
      if (sMask[m])
        Rpad[((size_t)st * NB + (r0 + m)) * RSTRIDE + k] =
            (k < H) ? sRNN[(st * 16 + m) * 52 + k] : 0.f;
    }
    if (lane < 16 && sMask[lane]) {
      float s = 0.f;
      for (int k = 0; k < H; ++k) {
        const float h = sRNN[(st * 16 + lane) * 52 + k];
        s += h * h;
      }
      rnrm[st * NB + (r0 + lane)] = s;
    }
  }
}

// ---------------------------------------------------------------------------
// Stage 2a: M = sqrt(clip(||Xi||^2 - 2 Xi.Yj + ||Yj||^2)) via WMMA Gram,
// plus grid-wide max/sum reduction for delta & mean.
// Block = 8 waves, each wave one 16x16 output tile.
// ---------------------------------------------------------------------------
__global__ __launch_bounds__(256) void gram_kernel(
    const float* __restrict__ Rt, const float* __restrict__ rnp,
    const float* __restrict__ tb, const float* __restrict__ t2b,
    float* __restrict__ M, float* __restrict__ scal) {
  const int lane = threadIdx.x & 31;
  const int wv = threadIdx.x >> 5;
  const int j0 = blockIdx.x * 16;
  const int i0 = (blockIdx.y * 8 + wv) * 16;
  const int half = lane >> 4, lm = lane & 15;

  v8f c = {0.f, 0.f, 0.f, 0.f, 0.f, 0.f, 0.f, 0.f};
  const float* arow = Rt + (size_t)(i0 + lm) * RSTRIDE + 2 * half;
  const float* brow = Rt + (size_t)(j0 + lm) * RSTRIDE + 2 * half;
#pragma unroll
  for (int kc = 0; kc < 13; ++kc) {
    v2f a = *(const v2f*)(arow + kc * 4);
    v2f b = *(const v2f*)(brow + kc * 4);
    c = wmma4(a, b, c);
  }
  const int j = j0 + lm;
  const float rnj = rnp[j];
  const float t2j = t2b[j];
  float lmax = 0.f, lsum = 0.f;
#pragma unroll
  for (int r = 0; r < 8; ++r) {
    const int i = i0 + r + 8 * half;
    const float ti = tb[i];
    const float sq =
        rnp[i] + ti * ti + rnj + t2j * t2j - 2.f * c[r] - 2.f * ti * t2j;
    const float m = sqrtf(fmaxf(sq, 1e-10f));
    M[(size_t)i * LDM + j] = m;
    lmax = fmaxf(lmax, m);
    lsum += m;
  }
  lmax = wave_max(lmax);
  lsum = wave_sum(lsum);
  if (lane == 0) {
    atomicMax((unsigned int*)scal, __float_as_uint(lmax));  // M >= 0
    atomicAdd(scal + 1, lsum);
  }
}

// ---------------------------------------------------------------------------
// Small setup kernels
// ---------------------------------------------------------------------------
__global__ void prep_kernel(const float* __restrict__ x,
                            const int* __restrict__ sidx,
                            float* __restrict__ tb, float* __restrict__ t2b,
                            float* __restrict__ dout) {
  const int i = blockIdx.x * 256 + threadIdx.x;
  if (i < NB) {
    tb[i] = x[(size_t)i * (H + 1)];
    t2b[i] = x[(size_t)sidx[i] * (H + 1)];
  }
  if (i == 0) dout[0] = 0.f;
}

__global__ void ts_init_kernel(float* __restrict__ scal) {
  scal[0] = 0.f;  // max bits
  scal[1] = 0.f;  // sum
  // CDNA5 wave-cluster split barrier (signal -3 / wait -3); architectural NOP
  // when ClusterID==0 (non-cluster dispatch), so safe here.
  __builtin_amdgcn_s_cluster_barrier();
}

__global__ void sink_init_kernel(float* __restrict__ u,
                                 float* __restrict__ scal) {
  const int i = blockIdx.x * 256 + threadIdx.x;
  if (i <= NB) u[i] = (i < NB) ? (0.5f / NB) : 0.5f;  // a vector (p=0.5)
  if (i == 0)
    scal[2] = 10.f * ((float)NB * (float)NB) / scal[1];  // eff_lam = lam/mean
}

// ---------------------------------------------------------------------------
// Sinkhorn matvecs.  K is never materialized: K = exp(-lam*M)+1e-6 computed
// on the fly (M stays resident in the 192MB L2).  Augmented row/col analytic.
// kt: v_j = b_j / (K^T u)_j     (column walk, coalesced across threads)
// kv: u_i = a_i / (K v)_i       (one wave per row)
// ---------------------------------------------------------------------------
__global__ __launch_bounds__(256) void kt_kernel(
    const float* __restrict__ M, const float* __restrict__ u,
    float* __restrict__ v, const float* __restrict__ scal) {
  const int j = blockIdx.x * 256 + threadIdx.x;
  if (j > NB) return;
  const float lam = scal[2];
  const float delta = scal[0];
  const float Kd = __expf(-lam * delta) + 1e-6f;
  const float K0 = 1.f + 1e-6f;
  float w;
  if (j < NB) {
    float acc = 0.f;
    for (int i = 0; i < NB; ++i) {
      if ((i & 15) == 0)
        __builtin_prefetch(&M[(size_t)(i + 16) * LDM + j], 0, 1);
      const float mij = M[(size_t)i * LDM + j];
      acc = fmaf(__expf(-lam * mij) + 1e-6f, u[i], acc);
    }
    w = acc + Kd * u[NB];
  } else {
    float acc = 0.f;
    for (int i = 0; i < NB; ++i) acc += u[i];
    w = Kd * acc + K0 * u[NB];
  }
  const float b = (j < NB) ? (0.5f / NB) : 0.5f;
  v[j] = b / w;
}

__global__ __launch_bounds__(256) void kv_kernel(
    const float* __restrict__ M, const float* __restrict__ v,
    float* __restrict__ u, const float* __restrict__ scal) {
  const int lane = threadIdx.x & 31;
  const int row = blockIdx.x * 8 + (threadIdx.x >> 5);
  if (row > NB) return;
  const float lam = scal[2];
  const float delta = scal[0];
  const float Kd = __expf(-lam * delta) + 1e-6f;
  const float K0 = 1.f + 1e-6f;
  float acc = 0.f;
  if (row < NB) {
    const float* mr = M + (size_t)row * LDM;
    for (int jj = lane; jj < NB; jj += 32)
      acc = fmaf(__expf(-lam * mr[jj]) + 1e-6f, v[jj], acc);
  } else {
    for (int jj = lane; jj < NB; jj += 32) acc += v[jj];
    acc *= Kd;
  }
  acc = wave_sum(acc);
  if (lane == 0) {
    const float r = (row < NB) ? (acc + Kd * v[NB]) : (acc + K0 * v[NB]);
    const float a = (row < NB) ? (0.5f / NB) : 0.5f;
    u[row] = a / r;
  }
}

// ---------------------------------------------------------------------------
// wass += 2 * sum_ij u_i K_ij v_j Mt_ij  (interior + analytic edges)
// ---------------------------------------------------------------------------
__global__ __launch_bounds__(256) void wass_row_kernel(
    const float* __restrict__ M, const float* __restrict__ u,
    const float* __restrict__ v, const float* __restrict__ scal,
    float* __restrict__ dout) {
  const int lane = threadIdx.x & 31;
  const int row = blockIdx.x * 8 + (threadIdx.x >> 5);
  if (row >= NB) return;
  const float lam = scal[2];
  const float* mr = M + (size_t)row * LDM;
  float acc = 0.f;
  for (int jj = lane; jj < NB; jj += 32) {
    const float m = mr[jj];
    acc = fmaf((__expf(-lam * m) + 1e-6f) * v[jj], m, acc);
  }
  acc = wave_sum(acc);
  if (lane == 0) atomicAdd(dout, 2.f * u[row] * acc);
}

__global__ void wass_edge_kernel(const float* __restrict__ u,
                                 const float* __restrict__ v,
                                 const float* __restrict__ scal,
                                 float* __restrict__ dout) {
  const int lane = threadIdx.x;
  float su = 0.f, sv = 0.f;
  for (int i = lane; i < NB; i += 32) {
    su += u[i];
    sv += v[i];
  }
  su = wave_sum(su);
  sv = wave_sum(sv);
  if (lane == 0) {
    const float lam = scal[2], delta = scal[0];
    const float Kd = __expf(-lam * delta) + 1e-6f;
    const float edge = v[NB] * Kd * delta * su + u[NB] * Kd * delta * sv;
    atomicAdd(dout, 2.f * edge);
  }
}

// ---------------------------------------------------------------------------
extern "C" void kernel_launch(void* const* d_in, const int* in_sizes, int n_in,
                              void* d_out, int out_size, void* d_ws,
                              size_t ws_size, hipStream_t stream) {
  const float* x = (const float*)d_in[0];
  const float* W_ih = (const float*)d_in[1];
  const float* W_hh = (const float*)d_in[2];
  const float* b_ih = (const float*)d_in[3];
  const float* b_hh = (const float*)d_in[4];
  const float* Wfc = (const float*)d_in[5];
  const float* bfc = (const float*)d_in[6];
  const float* u_att = (const float*)d_in[7];
  const float* Wt = (const float*)d_in[8];
  const float* bt = (const float*)d_in[9];
  const float* Ws1 = (const float*)d_in[10];
  const float* bs1 = (const float*)d_in[11];
  const float* Ws2 = (const float*)d_in[12];
  const float* bs2 = (const float*)d_in[13];
  const float* Wl1 = (const float*)d_in[14];
  const float* bl1 = (const float*)d_in[15];
  const float* Wl2 = (const float*)d_in[16];
  const float* bl2 = (const float*)d_in[17];
  const int* sidx = (const int*)d_in[18];

  float* dout = (float*)d_out;
  float* out_s = dout + 1;
  float* out_y = dout + 1 + (size_t)NB * TSN;

  // workspace carve (floats)
  float* ws = (float*)d_ws;
  size_t off = 0;
  float* Rpad = ws + off; off += (size_t)TSN * NB * RSTRIDE;  // 1.83M
  float* rnrm = ws + off; off += (size_t)TSN * NB;
  float* tb = ws + off;   off += NB;
  float* t2b = ws + off;  off += NB;
  float* u = ws + off;    off += 4104;
  float* v = ws + off;    off += 4104;
  float* scal = ws + off; off += 16;
  float* WihP = ws + off; off += (size_t)NE * WPAD;
  float* WhhP = ws + off; off += (size_t)NE * WPAD;
  float* bihP = ws + off; off += NE * 160;
  float* bhhP = ws + off; off += NE * 160;
  float* M = ws + off;    off += (size_t)NB * LDM;            // 16.8M (67MB)

  prep_kernel<<<(NB + 255) / 256, 256, 0, stream>>>(x, sidx, tb, t2b, dout);
  pad_w_kernel<<<(NE * WPAD + 255) / 256, 256, 0, stream>>>(
      W_ih, W_hh, b_ih, b_hh, WihP, WhhP, bihP, bhhP);

  expert_kernel<<<dim3(NB / 16, NE), 32, 0, stream>>>(
      x, WihP, WhhP, bihP, bhhP, Wfc, bfc, u_att, Wt, bt, Ws1, bs1, Ws2, bs2,
      Wl1, bl1, Wl2, bl2, Rpad, rnrm, out_s, out_y);

  for (int ts = 0; ts < TSN; ++ts) {
    const float* Rt = Rpad + (size_t)ts * NB * RSTRIDE;
    const float* rnp = rnrm + (size_t)ts * NB;
    ts_init_kernel<<<1, 1, 0, stream>>>(scal);
    gram_kernel<<<dim3(NB / 16, NB / 128), 256, 0, stream>>>(Rt, rnp, tb, t2b,
                                                             M, scal);
    sink_init_kernel<<<(NB + 256) / 256 + 1, 256, 0, stream>>>(u, scal);
    for (int it = 0; it < 10; ++it) {
      kt_kernel<<<(NB + 256) / 256 + 1, 256, 0, stream>>>(M, u, v, scal);
      kv_kernel<<<(NB + 8) / 8 + 1, 256, 0, stream>>>(M, v, u, scal);
    }
    kt_kernel<<<(NB + 256) / 256 + 1, 256, 0, stream>>>(M, u, v, scal);
    wass_row_kernel<<<NB / 8, 256, 0, stream>>>(M, u, v, scal, dout);
    wass_edge_kernel<<<1, 32, 0, stream>>>(u, v, scal, dout);
  }
}